// AuxiliaryGCN_84670985273383
// MI455X (gfx1250) — compile-verified
//
#include <hip/hip_runtime.h>
#include <math.h>

typedef float v2f __attribute__((ext_vector_type(2)));
typedef float v8f __attribute__((ext_vector_type(8)));

#define N_NODES 100000
#define FDIM 128
#define BN_EPS 1e-5f
#define XS 130    // LDS stride for x tile (pads away bank conflicts)
#define WROW 132  // LDS stride for W tile
#define MT 5      // M-tiles (of 16 rows) per block -> 80 rows; 100000 = 80*1250

// Guaranteed-native f32 atomic add (no CAS loop), device scope -> RMW in L2.
__device__ __forceinline__ void atomAddF(float* p, float v) {
  asm volatile("global_atomic_add_f32 %0, %1, off scope:SCOPE_DEV"
               :: "v"(p), "v"(v) : "memory");
}

// ---------- degree / normalization ----------
__global__ __launch_bounds__(256) void k_deg_init(float* deg) {
  int i = blockIdx.x * 256 + threadIdx.x;
  if (i < N_NODES) deg[i] = 1.0f;  // self loop
}
__global__ __launch_bounds__(256) void k_deg_acc(const int* __restrict__ col, int E, float* deg) {
  int e = blockIdx.x * 256 + threadIdx.x;
  if (e < E) atomAddF(&deg[col[e]], 1.0f);
}
__global__ __launch_bounds__(256) void k_dinv(float* deg) {
  int i = blockIdx.x * 256 + threadIdx.x;
  if (i < N_NODES) deg[i] = rsqrtf(deg[i]);   // deg >= 1 always
}

// ---------- fp32 WMMA GEMM: H = X @ W (128x128 W), also OutInit = H * dinv^2 ----------
// Block: 256 thr = 8 waves; block does 80 rows (5 M-tiles), wave w owns cols [16w,16w+16).
// One B fragment feeds 5 WMMAs (5 accumulators) -> 5x less W re-read from L2.
__global__ __launch_bounds__(256)
void k_gemm_wmma(const float* __restrict__ X, const float* __restrict__ W,
                 const float* __restrict__ dinv,
                 float* __restrict__ H, float* __restrict__ OutInit) {
  __shared__ float wls[128 * WROW];
  __shared__ float xls[16 * MT * XS];
  __shared__ float dl[16 * MT];
  const int tid = threadIdx.x;
  const int row0 = blockIdx.x * (16 * MT);

  for (int u = tid; u < 128 * 32; u += 256) {          // W: 128x128 as float4 units
    int r = u >> 5, c4 = (u & 31) * 4;
    const float4 v = *(const float4*)(W + r * 128 + c4);
    float* d = wls + r * WROW + c4;
    d[0] = v.x; d[1] = v.y; d[2] = v.z; d[3] = v.w;
  }
  for (int u = tid; u < 16 * MT * 32; u += 256) {      // X tile: 80x128
    int r = u >> 5, c4 = (u & 31) * 4;
    const float4 v = *(const float4*)(X + (size_t)(row0 + r) * FDIM + c4);
    float* d = xls + r * XS + c4;
    d[0] = v.x; d[1] = v.y; d[2] = v.z; d[3] = v.w;
  }
  if (tid < 16 * MT) dl[tid] = dinv[row0 + tid];
  __syncthreads();

  const int wave = tid >> 5, lane = tid & 31;
  const int hi = lane >> 4, n = lane & 15;
  const int colbase = wave * 16;

  v8f acc[MT];
#pragma unroll
  for (int t = 0; t < MT; ++t) acc[t] = (v8f){0.f,0.f,0.f,0.f,0.f,0.f,0.f,0.f};

#pragma unroll
  for (int k0 = 0; k0 < 128; k0 += 4) {
    // B frag (4x16 fp32): VGPR0 = K (lo lanes) / K+2 (hi lanes); VGPR1 = K+1 / K+3
    const float* wp = wls + (k0 + 2 * hi) * WROW + colbase + n;
    v2f b; b.x = wp[0]; b.y = wp[WROW];
#pragma unroll
    for (int t = 0; t < MT; ++t) {
      // A frag (16x4 fp32): lane lo -> (M=n, K=k0,k0+1); lane hi -> (M=n, K=k0+2,k0+3)
      const float* xp = xls + (t * 16 + n) * XS + k0 + 2 * hi;
      v2f a; a.x = xp[0]; a.y = xp[1];
      acc[t] = __builtin_amdgcn_wmma_f32_16x16x4_f32(false, a, false, b, (short)0,
                                                     acc[t], false, false);
    }
  }
#pragma unroll
  for (int t = 0; t < MT; ++t) {
#pragma unroll
    for (int v = 0; v < 8; ++v) {       // C/D: VGPR v -> M = v + 8*hi, N = n
      int m = t * 16 + v + 8 * hi;
      size_t idx = (size_t)(row0 + m) * FDIM + colbase + n;
      float d = dl[m];
      H[idx] = acc[t][v];
      OutInit[idx] = acc[t][v] * d * d;  // self-loop contribution
    }
  }
}

// ---------- edge scatter: wave per edge, float4 per lane, native f32 atomics ----------
__global__ __launch_bounds__(256)
void k_edge_agg(const int* __restrict__ row, const int* __restrict__ col, int E,
                const float* __restrict__ dinv, const float* __restrict__ H,
                float* __restrict__ Out) {
  int g = blockIdx.x * 256 + threadIdx.x;
  int e = g >> 5, lane = g & 31;
  if (e >= E) return;
  int r = row[e], c = col[e];
  float w = dinv[r] * dinv[c];
  const float4 v = *(const float4*)(H + (size_t)r * FDIM + lane * 4);
  float* o = Out + (size_t)c * FDIM + lane * 4;
  atomAddF(o + 0, v.x * w);
  atomAddF(o + 1, v.y * w);
  atomAddF(o + 2, v.z * w);
  atomAddF(o + 3, v.w * w);
}

// ---------- +b1 -> BN(eval) -> ReLU ----------
__global__ __launch_bounds__(256)
void k_bn_relu(const float* __restrict__ In, const float* __restrict__ b1,
               const float* __restrict__ gamma, const float* __restrict__ beta,
               const float* __restrict__ mean, const float* __restrict__ var,
               float* __restrict__ Out) {
  size_t i = (size_t)blockIdx.x * 256 + threadIdx.x;  // float4 index; exact grid
  int f0 = ((int)(i & 31)) * 4;
  float4 v = ((const float4*)In)[i];
  float in4[4] = {v.x, v.y, v.z, v.w};
  float4 rr;
  float* pr = &rr.x;
#pragma unroll
  for (int j = 0; j < 4; ++j) {
    int f = f0 + j;
    float sc = gamma[f] * rsqrtf(var[f] + BN_EPS);
    float val = (in4[j] + b1[f] - mean[f]) * sc + beta[f];
    pr[j] = fmaxf(val, 0.0f);
  }
  ((float4*)Out)[i] = rr;
}

// ---------- heads: wave per node, packed weight matrix in LDS ----------
// columns: 0..63 cls, 64..103 sim, 104 hom, 105 ent (padded to 128 with zeros)
__global__ __launch_bounds__(256)
void k_heads(const float* __restrict__ H2, const float* __restrict__ b2,
             const float* __restrict__ cls_w, const float* __restrict__ cls_b,
             const float* __restrict__ sim_w, const float* __restrict__ sim_b,
             const float* __restrict__ hom_w, const float* __restrict__ hom_b,
             const float* __restrict__ ent_w, const float* __restrict__ ent_b,
             float* __restrict__ out) {
  __shared__ float wall[128 * 128];
  __shared__ float h2s[8 * 128];
  __shared__ float bias_s[128];
  const int tid = threadIdx.x;
  for (int i = tid; i < 128 * 128; i += 256) {
    int f = i >> 7, c = i & 127;
    float w = 0.f;
    if (c < 64) w = cls_w[f * 64 + c];
    else if (c < 104) w = sim_w[f * 40 + (c - 64)];
    else if (c == 104) w = hom_w[f];
    else if (c == 105) w = ent_w[f];
    wall[i] = w;
  }
  if (tid < 128) {
    int c = tid;
    bias_s[c] = (c < 64) ? cls_b[c]
              : (c < 104) ? sim_b[c - 64]
              : (c == 104) ? hom_b[0]
              : (c == 105) ? ent_b[0] : 0.f;
  }
  __syncthreads();

  const int wave = tid >> 5, lane = tid & 31;
  float* h2 = h2s + wave * 128;
  const size_t simBase = (size_t)N_NODES * 64;
  const size_t homBase = simBase + (size_t)N_NODES * 40;
  const size_t entBase = homBase + (size_t)N_NODES;

  for (int it = 0; it < 8; ++it) {
    int node = blockIdx.x * 64 + wave * 8 + it;
    if (node >= N_NODES) return;           // uniform within wave
    float4 hv = *(const float4*)(H2 + (size_t)node * FDIM + lane * 4);
    const float4 bb = *(const float4*)(b2 + lane * 4);
    h2[lane * 4 + 0] = hv.x + bb.x;
    h2[lane * 4 + 1] = hv.y + bb.y;
    h2[lane * 4 + 2] = hv.z + bb.z;
    h2[lane * 4 + 3] = hv.w + bb.w;

    float a0 = 0.f, a1 = 0.f, a2 = 0.f, a3 = 0.f;
#pragma unroll 8
    for (int f = 0; f < 128; ++f) {
      float h = h2[f];
      const float* wr = wall + f * 128;
      a0 = fmaf(h, wr[lane], a0);
      a1 = fmaf(h, wr[lane + 32], a1);
      a2 = fmaf(h, wr[lane + 64], a2);
      a3 = fmaf(h, wr[lane + 96], a3);
    }
    a0 += bias_s[lane];
    a1 += bias_s[lane + 32];
    a2 += bias_s[lane + 64];
    a3 += bias_s[lane + 96];

    // log_softmax over 64 (lane -> c=lane, c=lane+32)
    float m = fmaxf(a0, a1);
#pragma unroll
    for (int off = 16; off; off >>= 1) m = fmaxf(m, __shfl_xor(m, off, 32));
    float s = __expf(a0 - m) + __expf(a1 - m);
#pragma unroll
    for (int off = 16; off; off >>= 1) s += __shfl_xor(s, off, 32);
    float L = m + __logf(s);
    out[(size_t)node * 64 + lane] = a0 - L;
    out[(size_t)node * 64 + 32 + lane] = a1 - L;

    // softmax over 40 (c=64+lane for all lanes, c=96+lane for lane<8)
    float v3 = (lane < 8) ? a3 : -3.0e38f;
    float ms = fmaxf(a2, v3);
#pragma unroll
    for (int off = 16; off; off >>= 1) ms = fmaxf(ms, __shfl_xor(ms, off, 32));
    float e2 = __expf(a2 - ms);
    float e3 = (lane < 8) ? __expf(v3 - ms) : 0.f;
    float ss = e2 + e3;
#pragma unroll
    for (int off = 16; off; off >>= 1) ss += __shfl_xor(ss, off, 32);
    float inv = 1.0f / ss;
    out[simBase + (size_t)node * 40 + lane] = e2 * inv;
    if (lane < 8) out[simBase + (size_t)node * 40 + 32 + lane] = e3 * inv;
    if (lane == 8) out[homBase + node] = 1.0f / (1.0f + __expf(-a3));  // c=104
    if (lane == 9) out[entBase + node] = 1.0f / (1.0f + __expf(-a3));  // c=105
  }
}

extern "C" void kernel_launch(void* const* d_in, const int* in_sizes, int n_in,
                              void* d_out, int out_size, void* d_ws, size_t ws_size,
                              hipStream_t stream) {
  const float* x = (const float*)d_in[0];
  const int* ei = (const int*)d_in[1];
  const float* w1 = (const float*)d_in[2];
  const float* b1 = (const float*)d_in[3];
  const float* w2 = (const float*)d_in[4];
  const float* b2 = (const float*)d_in[5];
  const float* bn_g = (const float*)d_in[6];
  const float* bn_b = (const float*)d_in[7];
  const float* bn_m = (const float*)d_in[8];
  const float* bn_v = (const float*)d_in[9];
  const float* cls_w = (const float*)d_in[10];
  const float* cls_b = (const float*)d_in[11];
  const float* sim_w = (const float*)d_in[12];
  const float* sim_b = (const float*)d_in[13];
  const float* hom_w = (const float*)d_in[14];
  const float* hom_b = (const float*)d_in[15];
  const float* ent_w = (const float*)d_in[16];
  const float* ent_b = (const float*)d_in[17];
  float* out = (float*)d_out;

  const int E = in_sizes[1] / 2;
  const int* rowi = ei;
  const int* coli = ei + E;

  // workspace layout (bytes): dinv | A | B | C   (~155 MB total)
  const size_t NB = (size_t)N_NODES * FDIM * sizeof(float);  // 51.2 MB
  char* wsb = (char*)d_ws;
  float* dinv = (float*)wsb;
  float* A = (float*)(wsb + (1u << 20));
  float* B = (float*)(wsb + (1u << 20) + NB);
  float* C = (float*)(wsb + (1u << 20) + 2 * NB);

  const int nodeBlocks = (N_NODES + 255) / 256;          // 391
  const int edgeBlocks = (E + 255) / 256;                // 6250
  const int waveEdgeBlocks = (int)(((size_t)E * 32 + 255) / 256);  // 200000
  const int gemmBlocks = N_NODES / (16 * MT);            // 1250 exact
  const int bnBlocks = (int)((size_t)N_NODES * FDIM / 4 / 256);    // 12500 exact
  const int headBlocks = (N_NODES + 63) / 64;            // 1563

  // normalization
  k_deg_init<<<nodeBlocks, 256, 0, stream>>>(dinv);
  k_deg_acc<<<edgeBlocks, 256, 0, stream>>>(coli, E, dinv);
  k_dinv<<<nodeBlocks, 256, 0, stream>>>(dinv);

  // layer 1
  k_gemm_wmma<<<gemmBlocks, 256, 0, stream>>>(x, w1, dinv, A, B);   // A=hx1, B=self-loop init
  k_edge_agg<<<waveEdgeBlocks, 256, 0, stream>>>(rowi, coli, E, dinv, A, B);
  k_bn_relu<<<bnBlocks, 256, 0, stream>>>(B, b1, bn_g, bn_b, bn_m, bn_v, A);  // A=h1_act

  // layer 2
  k_gemm_wmma<<<gemmBlocks, 256, 0, stream>>>(A, w2, dinv, B, C);   // B=hx2, C=init
  k_edge_agg<<<waveEdgeBlocks, 256, 0, stream>>>(rowi, coli, E, dinv, B, C);

  // heads -> d_out (main 6.4M | sim 4.0M | hom 0.1M | ent 0.1M floats)
  k_heads<<<headBlocks, 256, 0, stream>>>(C, b2, cls_w, cls_b, sim_w, sim_b,
                                          hom_w, hom_b, ent_w, ent_b, out);
  (void)n_in; (void)out_size; (void)ws_size;
}